// Attention_15375982920258
// MI455X (gfx1250) — compile-verified
//
#include <hip/hip_runtime.h>
#include <math.h>

// ---------------------------------------------------------------------------
// Problem constants (from the reference):
//   B=64, K=16, Q=2048, DIM=1024
// d_out = [ out (64*1024) | attn (64*16*2048) ]  (fp32)
//
// Roofline: dominant traffic is query (512 MB fp32) -> memory bound
// (~22us at 23.3 TB/s). Kernel 1 fuses scores+softmax+energy so query is
// read exactly once, streamed into LDS by the Tensor Data Mover with
// double buffering on TENSORcnt. All matmuls use V_WMMA_F32_16X16X4_F32.
// ---------------------------------------------------------------------------
#define BB      64
#define KK      16
#define QQ      2048
#define DIMD    1024
#define QSPLIT  4
#define QCH     (QQ / QSPLIT)   // 512 queries per workgroup

typedef float v2f __attribute__((ext_vector_type(2)));
typedef float v8f __attribute__((ext_vector_type(8)));
typedef unsigned int u32x4 __attribute__((ext_vector_type(4)));
typedef int i32x4 __attribute__((ext_vector_type(4)));
typedef int i32x8 __attribute__((ext_vector_type(8)));

// V_WMMA_F32_16X16X4_F32 : D(16x16 f32) = A(16x4 f32) * B(4x16 f32) + C
//   A: lane L holds row m=L&15, K = {2*(L>>4), 2*(L>>4)+1}   -> v2f
//   B: lane L holds col n=L&15, K = {2*(L>>4), 2*(L>>4)+1}   -> v2f
//   C/D: VGPR j, lanes 0-15 -> D[j][lane]; lanes 16-31 -> D[8+j][lane-16]
__device__ __forceinline__ v8f wmma_f32(v2f a, v2f b, v8f c) {
  return __builtin_amdgcn_wmma_f32_16x16x4_f32(false, a, false, b,
                                               (short)0, c, false, false);
}

// 8-byte aligned v2f load (lowers to ds_load_b64 / global_load_b64).
__device__ __forceinline__ v2f ld2(const float* p) { return *(const v2f*)p; }

// ---------------------------------------------------------------------------
// Tensor Data Mover (gfx1250): 2-D tile DMA global->LDS, HW LDS padding:
//  - 1024-dword rows, +1 dword per 256 dwords  -> row stride 1028 (4-bank rot)
//  - 128-dword rows,  +4 dwords per 128 dwords -> row stride 132  (4-bank rot)
// Mandatory path: builtin if declared, else inline-asm (ISA mnemonic).
// ---------------------------------------------------------------------------
__device__ __forceinline__ void tdm_load_2d(const void* gsrc, void* ldst,
                                            unsigned tensor_d0, unsigned tensor_d1,
                                            unsigned tile_d0, unsigned tile_d1,
                                            unsigned long long stride0_elems,
                                            unsigned pad_interval_code,
                                            unsigned pad_amount_code) {
  const unsigned long long ga = (unsigned long long)(size_t)gsrc;
  const unsigned la = (unsigned)(size_t)ldst;  // LDS aperture: addr[31:0] = LDS offset
  u32x4 g0;
  g0[0] = 1u;                                   // count=1, user-mode descriptor
  g0[1] = la;                                   // lds_addr
  g0[2] = (unsigned)ga;                         // global_addr[31:0]
  g0[3] = (unsigned)((ga >> 32) & 0x01FFFFFFull) | (2u << 30);  // addr[56:32]|type=2
  i32x8 g1;
  g1[0] = (int)((2u << 16) |                    // data_size = 4B
                (1u << 20) |                    // pad_enable
                (pad_interval_code << 22) | (pad_amount_code << 25));
  g1[1] = (int)((tensor_d0 & 0xFFFFu) << 16);   // atomic_barrier=0 | tensor_dim0 lo
  g1[2] = (int)((tensor_d0 >> 16) | ((tensor_d1 & 0xFFFFu) << 16));
  g1[3] = (int)((tensor_d1 >> 16) | (tile_d0 << 16));
  g1[4] = (int)(tile_d1);                       // tile_dim1 | tile_dim2=0
  g1[5] = (int)(unsigned)(stride0_elems & 0xFFFFFFFFull);
  g1[6] = (int)(unsigned)((stride0_elems >> 32) & 0xFFFFull);  // | dim1_stride=0
  g1[7] = 0;
  i32x4 g2 = {0, 0, 0, 0};
  i32x4 g3 = {0, 0, 0, 0};
#if __has_builtin(__builtin_amdgcn_tensor_load_to_lds)
#if defined(__clang_major__) && (__clang_major__ >= 23)
  i32x8 gz = {0, 0, 0, 0, 0, 0, 0, 0};
  __builtin_amdgcn_tensor_load_to_lds(g0, g1, g2, g3, gz, 0);
#else
  __builtin_amdgcn_tensor_load_to_lds(g0, g1, g2, g3, 0);
#endif
#else
  asm volatile("tensor_load_to_lds %0, %1, %2, %3"
               :: "s"(g0), "s"(g1), "s"(g2), "s"(g3)
               : "memory");
#endif
}

__device__ __forceinline__ void wait_tensorcnt_le1() {
#if __has_builtin(__builtin_amdgcn_s_wait_tensorcnt)
  __builtin_amdgcn_s_wait_tensorcnt(1);
#else
  asm volatile("s_wait_tensorcnt 0x1" ::: "memory");
#endif
}
__device__ __forceinline__ void wait_tensorcnt_le2() {
#if __has_builtin(__builtin_amdgcn_s_wait_tensorcnt)
  __builtin_amdgcn_s_wait_tensorcnt(2);
#else
  asm volatile("s_wait_tensorcnt 0x2" ::: "memory");
#endif
}
__device__ __forceinline__ void wait_tensorcnt_0() {
#if __has_builtin(__builtin_amdgcn_s_wait_tensorcnt)
  __builtin_amdgcn_s_wait_tensorcnt(0);
#else
  asm volatile("s_wait_tensorcnt 0x0" ::: "memory");
#endif
}

// Padded index for 1024-dword rows with +1 dword per 256: stride 1028.
__device__ __forceinline__ int qpad(int r, int f) {
  return r * 1028 + f + (f >> 8);
}

#define TILE_F 16448   // 16 rows * 1028
#define SPSTR 18
#define PSTR  18
#define CSTR  2052
#define WSTR  132
#define ASTR  132

// ===========================================================================
// Kernel 1: fused scores -> softmax(axis=K) -> attn out -> partial energy.
// grid = (B, QSPLIT), block = 256 (8 waves). Query tiles double-buffered
// through the TDM; key tile loaded once.
// ===========================================================================
__global__ __launch_bounds__(256)
void attn_energy_kernel(const float* __restrict__ key,
                        const float* __restrict__ query,
                        float* __restrict__ attn_out,
                        float* __restrict__ epart) {
  extern __shared__ float lds[];
  float* keyT   = lds;                       // 16 x 1028
  float* qbuf0  = keyT + TILE_F;             // 16 x 1028
  float* qbuf1  = qbuf0 + TILE_F;            // 16 x 1028
  float* spart  = qbuf1 + TILE_F;            // 8 x 16 x SPSTR
  float* sfull  = spart + 8 * 16 * SPSTR;    // 16 x 16
  float* colmax = sfull + 256;               // 16
  float* colinv = colmax + 16;               // 16
  float* Pl     = colinv + 16;               // 16 x PSTR

  const int b    = blockIdx.x;
  const int qs   = blockIdx.y;
  const int tid  = threadIdx.x;
  const int wave = tid >> 5;
  const int lane = tid & 31;
  const int ln   = lane & 15;
  const int half = lane >> 4;
  const int q0base = qs * QCH;
  const int NIT = QCH / 16;

  if (wave == 0) {
    tdm_load_2d(key + (size_t)b * KK * DIMD, keyT, DIMD, KK, DIMD, KK,
                DIMD, 7, 0);
    tdm_load_2d(query + ((size_t)b * QQ + q0base) * DIMD, qbuf0, DIMD, KK,
                DIMD, KK, DIMD, 7, 0);
  }

  v8f eacc[8];
#pragma unroll
  for (int t = 0; t < 8; ++t)
    eacc[t] = (v8f){0.f, 0.f, 0.f, 0.f, 0.f, 0.f, 0.f, 0.f};

  for (int it = 0; it < NIT; ++it) {
    float* qT = (it & 1) ? qbuf1 : qbuf0;
    const int q0 = q0base + it * 16;

    if (wave == 0) {
      if (it + 1 < NIT) {
        tdm_load_2d(query + ((size_t)b * QQ + q0 + 16) * DIMD,
                    (it & 1) ? qbuf0 : qbuf1, DIMD, KK, DIMD, KK, DIMD, 7, 0);
        wait_tensorcnt_le1();   // tile 'it' (and key) complete
      } else {
        wait_tensorcnt_0();
      }
    }
    __syncthreads();

    // ---- partial scores: wave reduces DIM slice [wave*128, +128) ----
    v8f s = (v8f){0.f, 0.f, 0.f, 0.f, 0.f, 0.f, 0.f, 0.f};
    const int kdbase = wave * 128;
#pragma unroll 4
    for (int ks = 0; ks < 32; ++ks) {
      const int kd = kdbase + ks * 4 + 2 * half;
      const int ix = ln * 1028 + kd + (kd >> 8);   // may be odd -> scalar pair
      v2f a  = {keyT[ix], keyT[ix + 1]};
      v2f bb = {qT[ix], qT[ix + 1]};
      s = wmma_f32(a, bb, s);
    }
#pragma unroll
    for (int j = 0; j < 8; ++j)
      spart[(wave * 16 + (j + 8 * half)) * SPSTR + ln] = s[j];
    __syncthreads();

    // ---- reduce 8 partials; thread t -> (m = t>>4, n = t&15) ----
    {
      const int m = tid >> 4, n = tid & 15;
      float v = 0.f;
#pragma unroll
      for (int w = 0; w < 8; ++w) v += spart[(w * 16 + m) * SPSTR + n];
      sfull[m * 16 + n] = v;
    }
    __syncthreads();

    // ---- softmax over the K axis (16 values per query column) ----
    if (tid < 16) {
      const int n = tid;
      float mx = -3.0e38f;
      for (int m = 0; m < 16; ++m) mx = fmaxf(mx, sfull[m * 16 + n]);
      float sum = 0.f;
      for (int m = 0; m < 16; ++m) sum += __expf(sfull[m * 16 + n] - mx);
      colmax[n] = mx;
      colinv[n] = 1.f / sum;
    }
    __syncthreads();
    {
      const int m = tid >> 4, n = tid & 15;
      const float p = __expf(sfull[m * 16 + n] - colmax[n]) * colinv[n];
      Pl[m * PSTR + n] = p;
      attn_out[((size_t)b * KK + m) * QQ + q0 + n] = p;   // attn[b][k][q]
    }
    __syncthreads();

    // ---- energy: E[m][d] += sum_k P[m][k] * query[q0+k][d] ----
#pragma unroll
    for (int t = 0; t < 8; ++t) {
      const int d0 = wave * 128 + t * 16;
      const int dc = (d0 + ln) + ((d0 + ln) >> 8);   // padded column offset
      v8f acc = eacc[t];
#pragma unroll
      for (int k0 = 0; k0 < 16; k0 += 4) {
        const int kq = k0 + 2 * half;
        v2f a  = ld2(Pl + ln * PSTR + kq);           // even offset: ds_load_b64
        v2f bb = {qT[kq * 1028 + dc], qT[(kq + 1) * 1028 + dc]};
        acc = wmma_f32(a, bb, acc);
      }
      eacc[t] = acc;
    }
    __syncthreads();
  }

  // ---- write partial energy: epart[qs][b][m][d] ----
#pragma unroll
  for (int t = 0; t < 8; ++t) {
    const int d = wave * 128 + t * 16 + ln;
#pragma unroll
    for (int j = 0; j < 8; ++j) {
      const int m = j + 8 * half;
      epart[(((size_t)qs * BB + b) * KK + m) * DIMD + d] = eacc[t][j];
    }
  }
}

// ===========================================================================
// Kernel 2: hidden[b,k,d] = tanh( sum_f combind[b,k,f] * W_lin[d,f] + b_lin[d] )
// combind = [sum_qs epart | key], 16 x 2048 per batch, staged in LDS.
// grid = B, block = 256 (wave w owns d in [w*128, +128)). W_lin (8 MB) is
// L2-resident across the 64 WGs.
// ===========================================================================
__global__ __launch_bounds__(256)
void hidden_kernel(const float* __restrict__ key,
                   const float* __restrict__ epart,
                   const float* __restrict__ W_lin,
                   const float* __restrict__ b_lin,
                   float* __restrict__ hidden) {
  extern __shared__ float lds[];
  float* C = lds;  // 16 x CSTR

  const int b    = blockIdx.x;
  const int tid  = threadIdx.x;
  const int wave = tid >> 5;
  const int lane = tid & 31;
  const int ln   = lane & 15;
  const int half = lane >> 4;

  for (int i = tid; i < KK * DIMD; i += 256) {
    const int m = i >> 10, f = i & 1023;
    float v = 0.f;
#pragma unroll
    for (int q = 0; q < QSPLIT; ++q)
      v += epart[(((size_t)q * BB + b) * KK + m) * DIMD + f];
    C[m * CSTR + f] = v;
  }
  for (int i = tid; i < KK * (DIMD / 4); i += 256) {
    const int m = i >> 8, c4 = i & 255;
    const float4 v = ((const float4*)(key + ((size_t)b * KK + m) * DIMD))[c4];
    float* dst = C + m * CSTR + DIMD + c4 * 4;
    dst[0] = v.x; dst[1] = v.y; dst[2] = v.z; dst[3] = v.w;
  }
  __syncthreads();

  const float* crow = C + ln * CSTR;   // even stride: v2f loads are 8B aligned
#pragma unroll 2
  for (int t = 0; t < 8; ++t) {
    const int d0 = wave * 128 + t * 16;
    v8f acc = (v8f){0.f, 0.f, 0.f, 0.f, 0.f, 0.f, 0.f, 0.f};
    const float* wrowbase = W_lin + (size_t)(d0 + ln) * (2 * DIMD);
#pragma unroll 4
    for (int f0 = 0; f0 < 2 * DIMD; f0 += 4) {
      const int ff = f0 + 2 * half;
      v2f a  = ld2(crow + ff);          // ds_load_b64
      v2f bb = ld2(wrowbase + ff);      // global_load_b64
      acc = wmma_f32(a, bb, acc);
    }
    const int d = d0 + ln;
    const float bl = b_lin[d];
#pragma unroll
    for (int j = 0; j < 8; ++j) {
      const int m = j + 8 * half;
      hidden[((size_t)b * KK + m) * DIMD + d] = tanhf(acc[j] + bl);
    }
  }
}

// ===========================================================================
// Kernel 3: out[b,d] = sum_f hidden_flat[b,f] * W_fc[d,f] + b_fc[d]
// M=64, N=1024, K=16384. grid = 8 (128-column d-chunks), block = 256.
// W_fc (64 MB, dominant stream here) + hidden tiles double-buffered through
// the TDM; HW pad (+4 dw per 128 dw) gives row stride 132.
// ===========================================================================
#define KC3 128
__global__ __launch_bounds__(256)
void out_kernel(const float* __restrict__ hidden,
                const float* __restrict__ W_fc,
                const float* __restrict__ b_fc,
                float* __restrict__ out) {
  extern __shared__ float lds[];
  float* Wt0 = lds;                  // 128 x WSTR
  float* At0 = Wt0 + 128 * WSTR;     // 64 x ASTR
  float* Wt1 = At0 + 64 * ASTR;      // 128 x WSTR
  float* At1 = Wt1 + 128 * WSTR;     // 64 x ASTR

  const int dglob0 = blockIdx.x * 128;
  const int tid  = threadIdx.x;
  const int wave = tid >> 5;
  const int lane = tid & 31;
  const int ln   = lane & 15;
  const int half = lane >> 4;
  const int rt   = wave & 3;
  const int dsub = (wave >> 2) * 64;

  v8f acc[4];
#pragma unroll
  for (int t = 0; t < 4; ++t)
    acc[t] = (v8f){0.f, 0.f, 0.f, 0.f, 0.f, 0.f, 0.f, 0.f};

  const int KF = KK * DIMD;   // 16384
  const int NCH = KF / KC3;   // 128

  if (wave == 0) {
    tdm_load_2d(W_fc + (size_t)dglob0 * KF, Wt0, KC3, 128, KC3, 128, KF, 6, 3);
    tdm_load_2d(hidden, At0, KC3, 64, KC3, 64, KF, 6, 3);
  }

  for (int c = 0; c < NCH; ++c) {
    float* Wt = (c & 1) ? Wt1 : Wt0;
    float* At = (c & 1) ? At1 : At0;

    if (wave == 0) {
      if (c + 1 < NCH) {
        const int kg = (c + 1) * KC3;
        tdm_load_2d(W_fc + (size_t)dglob0 * KF + kg, (c & 1) ? Wt0 : Wt1,
                    KC3, 128, KC3, 128, KF, 6, 3);
        tdm_load_2d(hidden + kg, (c & 1) ? At0 : At1, KC3, 64, KC3, 64, KF,
                    6, 3);
        wait_tensorcnt_le2();   // chunk c complete
      } else {
        wait_tensorcnt_0();
      }
    }
    __syncthreads();

    const float* arow = At + (rt * 16 + ln) * ASTR;   // even strides ->
    const float* wrow0 = Wt + (dsub + ln) * WSTR;     // 8B-aligned v2f loads
#pragma unroll
    for (int t = 0; t < 4; ++t) {
      const float* wrow = wrow0 + t * 16 * WSTR;
      v8f a8 = acc[t];
#pragma unroll 8
      for (int k0 = 0; k0 < KC3; k0 += 4) {
        const int kk = k0 + 2 * half;
        v2f a  = ld2(arow + kk);    // ds_load_b64
        v2f bb = ld2(wrow + kk);    // ds_load_b64
        a8 = wmma_f32(a, bb, a8);
      }
      acc[t] = a8;
    }
    __syncthreads();
  }

#pragma unroll
  for (int t = 0; t < 4; ++t) {
    const int d = dglob0 + dsub + t * 16 + ln;
    const float bf = b_fc[d];
#pragma unroll
    for (int j = 0; j < 8; ++j) {
      const int brow = rt * 16 + j + 8 * half;
      out[(size_t)brow * DIMD + d] = acc[t][j] + bf;
    }
  }
}

// ===========================================================================
extern "C" void kernel_launch(void* const* d_in, const int* in_sizes, int n_in,
                              void* d_out, int out_size, void* d_ws,
                              size_t ws_size, hipStream_t stream) {
  (void)in_sizes; (void)n_in; (void)out_size; (void)ws_size;
  const float* key   = (const float*)d_in[0];
  const float* query = (const float*)d_in[1];
  const float* W_lin = (const float*)d_in[2];
  const float* b_lin = (const float*)d_in[3];
  const float* W_fc  = (const float*)d_in[4];
  const float* b_fc  = (const float*)d_in[5];

  float* out  = (float*)d_out;                 // [64, 1024]
  float* attn = out + (size_t)BB * DIMD;       // [64, 16, 2048]

  float* epart  = (float*)d_ws;                                   // 16 MB
  float* hidden = epart + (size_t)QSPLIT * BB * KK * DIMD;        // +4 MB

  const size_t lds1 =
      (size_t)(3 * TILE_F + 8 * 16 * SPSTR + 256 + 16 + 16 + 16 * PSTR) *
      sizeof(float);                                  // ~204 KB
  const size_t lds2 = (size_t)(16 * CSTR) * sizeof(float);        // ~128 KB
  const size_t lds3 =
      (size_t)(2 * (128 * WSTR + 64 * ASTR)) * sizeof(float);     // ~198 KB

  (void)hipFuncSetAttribute((const void*)attn_energy_kernel,
                            hipFuncAttributeMaxDynamicSharedMemorySize,
                            (int)lds1);
  (void)hipFuncSetAttribute((const void*)hidden_kernel,
                            hipFuncAttributeMaxDynamicSharedMemorySize,
                            (int)lds2);
  (void)hipFuncSetAttribute((const void*)out_kernel,
                            hipFuncAttributeMaxDynamicSharedMemorySize,
                            (int)lds3);

  attn_energy_kernel<<<dim3(BB, QSPLIT), 256, lds1, stream>>>(key, query, attn,
                                                              epart);
  hidden_kernel<<<dim3(BB), 256, lds2, stream>>>(key, epart, W_lin, b_lin,
                                                 hidden);
  out_kernel<<<dim3(DIMD / 128), 256, lds3, stream>>>(hidden, W_fc, b_fc, out);
}